// GaussExpFalloffKernel_83434034692735
// MI455X (gfx1250) — compile-verified
//
#include <hip/hip_runtime.h>

typedef __attribute__((ext_vector_type(2))) float v2f;
typedef __attribute__((ext_vector_type(8))) float v8f;

#define LOG2E 1.44269504088896340736f

// Precompute per-vortex constants (stride 8 floats = 32B):
// [ -2y, -2x, y^2+x^2, tau, -log2e/sig^2, -log2e/(d*sig^2), c, c*log2e ]
__global__ __launch_bounds__(256) void vortex_prep(const float* __restrict__ vf,
                                                   float* __restrict__ ws, int n) {
  int i = blockIdx.x * blockDim.x + threadIdx.x;
  if (i >= n) return;
  float y   = vf[i * 6 + 0];
  float x   = vf[i * 6 + 1];
  float tau = vf[i * 6 + 2];
  float sig = vf[i * 6 + 3];
  float c   = vf[i * 6 + 4];
  float d   = vf[i * 6 + 5];
  float a = -LOG2E / (sig * sig);      // gaussian exponent scale (base-2)
  float* o = ws + i * 8;
  o[0] = -2.0f * y;
  o[1] = -2.0f * x;
  o[2] = y * y + x * x;
  o[3] = tau;
  o[4] = a;
  o[5] = a / d;                        // DT = 1
  o[6] = c;
  o[7] = c * LOG2E;
}

// One wave32 handles 16 points x all P vortices.
// WMMA computes the 16x16 sq-distance tile on the matrix pipe:
//   A[p,:] = [py, px, pp, 1],  B[:,v] = [-2ly, -2lx, 1, ll]  ->  D[p,v] = |p-l|^2
// D layout: lanes 0-15 -> vortex N=lane, slots M=0..7 (points 0..7)
//           lanes 16-31 -> vortex N=lane-16, slots M=8..15 (points 8..15)
__global__ __launch_bounds__(256) void vortex_vel(const float* __restrict__ pts,
                                                  const float* __restrict__ ws,
                                                  float* __restrict__ out,
                                                  int ptsPerBatch, int P) {
  const int lane = threadIdx.x & 31;
  const int wave = blockIdx.x * (blockDim.x >> 5) + (threadIdx.x >> 5);
  const int pointBase = wave * 16;
  const int batch = pointBase / ptsPerBatch;
  const float* wsb = ws + (size_t)batch * P * 8;
  const bool hi = lane >= 16;
  const int p = lane & 15;

  // Build A operand once (lanes 0-15: K0,K1 = py,px ; lanes 16-31: K2,K3 = pp,1)
  float py = pts[2 * (pointBase + p) + 0];
  float px = pts[2 * (pointBase + p) + 1];
  float pp = py * py + px * px;
  v2f A;
  A.x = hi ? pp : py;
  A.y = hi ? 1.0f : px;

  float S[8], SX[8], SY[8];
#pragma unroll
  for (int v = 0; v < 8; ++v) { S[v] = 0.0f; SX[v] = 0.0f; SY[v] = 0.0f; }

  const int ntiles = P >> 4;
  for (int t = 0; t < ntiles; ++t) {
    const float* w = wsb + (size_t)(t * 16 + p) * 8;
    float m2y = w[0], m2x = w[1], ll = w[2], tauv = w[3];
    float a = w[4], b = w[5], c = w[6], cl2 = w[7];

    // B operand (lanes 0-15: K0,K1 = -2ly,-2lx ; lanes 16-31: K2,K3 = 1,ll)
    v2f Bm;
    Bm.x = hi ? 1.0f : m2y;
    Bm.y = hi ? ll   : m2x;

    v8f C = {};
    v8f D = __builtin_amdgcn_wmma_f32_16x16x4_f32(false, A, false, Bm,
                                                  (short)0, C, false, false);
    union { v8f v; float f[8]; } u;
    u.v = D;

#pragma unroll
    for (int v = 0; v < 8; ++v) {
      float sq = u.f[v];
      float g  = __builtin_amdgcn_exp2f(sq * a);          // exp(-sq/sig^2)
      float e1 = __builtin_amdgcn_exp2f(sq * b);          // exp(-sq/(d*sig^2))
      float ce = c * e1;
      float e2 = __builtin_amdgcn_exp2f(-(cl2 * e1));     // exp(-c*e1)
      float r  = __builtin_amdgcn_rsqf(sq + ce);          // 1/sqrt(sq + c*e1)
      float s  = tauv * g * e2 * r;
      S[v]  += s;
      SX[v]  = fmaf(s, m2x, SX[v]);                       // Σ s * (-2 lx)
      SY[v]  = fmaf(s, m2y, SY[v]);                       // Σ s * (-2 ly)
    }
  }

  // Reduce over the 16 vortex lanes within each half-wave (xor masks < 16
  // never cross the half boundary).
#pragma unroll
  for (int m = 1; m <= 8; m <<= 1) {
#pragma unroll
    for (int v = 0; v < 8; ++v) {
      S[v]  += __shfl_xor(S[v],  m, 32);
      SX[v] += __shfl_xor(SX[v], m, 32);
      SY[v] += __shfl_xor(SY[v], m, 32);
    }
  }

  // Lane 0 writes points 0..7, lane 16 writes points 8..15.
  if (lane == 0 || lane == 16) {
    int pb = pointBase + (hi ? 8 : 0);
#pragma unroll
    for (int v = 0; v < 8; ++v) {
      float qy = pts[2 * (pb + v) + 0];
      float qx = pts[2 * (pb + v) + 1];
      // vel_y = S*px - Σ s*lx = S*qx + 0.5*SX ; vel_x = -(S*py - Σ s*ly)
      float vy =  S[v] * qx + 0.5f * SX[v];
      float vx = -S[v] * qy - 0.5f * SY[v];
      out[2 * (pb + v) + 0] = vy;
      out[2 * (pb + v) + 1] = vx;
    }
  }
}

extern "C" void kernel_launch(void* const* d_in, const int* in_sizes, int n_in,
                              void* d_out, int out_size, void* d_ws, size_t ws_size,
                              hipStream_t stream) {
  const float* vf  = (const float*)d_in[0];   // (B, P, 6) f32
  const float* pts = (const float*)d_in[1];   // (B, H, W, 2) f32
  float* out = (float*)d_out;                 // (B, H, W, 2) f32
  float* ws  = (float*)d_ws;

  const int B = 2;
  const int nvort = in_sizes[0] / 6;          // B*P = 512
  const int P = nvort / B;                    // 256
  const int npts = in_sizes[1] / 2;           // B*H*W = 131072
  const int ptsPerBatch = npts / B;           // 65536

  vortex_prep<<<(nvort + 255) / 256, 256, 0, stream>>>(vf, ws, nvort);

  const int waves = npts / 16;                // 8192 waves
  vortex_vel<<<waves / 8, 256, 0, stream>>>(pts, ws, out, ptsPerBatch, P);
}